// TraceFIMbackup_90847148245671
// MI455X (gfx1250) — compile-verified
//
#include <hip/hip_runtime.h>
#include <math.h>

typedef __attribute__((ext_vector_type(2))) float v2f;
typedef __attribute__((ext_vector_type(8))) float v8f;

#define NB 128
#define ND 512
#define NH 1024
#define NC 1000
#define NCP 1024  // padded class dim

// ---------------- threefry2x32 (JAX-compatible, 20 rounds) ----------------
__device__ __forceinline__ void threefry2x32(unsigned k0, unsigned k1,
                                             unsigned x0, unsigned x1,
                                             unsigned& o0, unsigned& o1) {
  unsigned ks2 = k0 ^ k1 ^ 0x1BD11BDAu;
  x0 += k0; x1 += k1;
#define TFR(r) { x0 += x1; x1 = (x1 << (r)) | (x1 >> (32 - (r))); x1 ^= x0; }
  TFR(13) TFR(15) TFR(26) TFR(6)
  x0 += k1;  x1 += ks2 + 1u;
  TFR(17) TFR(29) TFR(16) TFR(24)
  x0 += ks2; x1 += k0 + 2u;
  TFR(13) TFR(15) TFR(26) TFR(6)
  x0 += k0;  x1 += k1 + 3u;
  TFR(17) TFR(29) TFR(16) TFR(24)
  x0 += k1;  x1 += ks2 + 4u;
  TFR(13) TFR(15) TFR(26) TFR(6)
  x0 += ks2; x1 += k0 + 5u;
#undef TFR
  o0 = x0; o1 = x1;
}

__device__ __forceinline__ float gumbel_from_bits(unsigned bits) {
  // JAX uniform: f = bitcast((bits>>9)|0x3f800000)-1 in [0,1); u in [tiny,1)
  float f = __uint_as_float((bits >> 9) | 0x3f800000u) - 1.0f;
  const float tiny = 1.1754943508222875e-38f;
  float u = f * (1.0f - tiny) + tiny;
  u = fmaxf(u, tiny);
  return -logf(-logf(u));
}

// ---------------- GEMM1: h = relu(x @ W1 + b1)   [128x512 @ 512x1024] -----
// 128 waves total: wave = (mt in 0..7) * 16 + (ng in 0..15); each wave does
// one 16-row M tile x four 16-col N tiles. fp32 WMMA 16x16x4.
__global__ void k_gemm1(const float* __restrict__ x, const float* __restrict__ W1,
                        const float* __restrict__ bias1, float* __restrict__ h) {
  int w    = blockIdx.x * (blockDim.x >> 5) + (threadIdx.x >> 5);
  int lane = threadIdx.x & 31;
  int kh   = lane >> 4;      // lane half: selects K offset +0 / +2
  int l16  = lane & 15;
  int mt = w >> 4, ng = w & 15;
  int m = mt * 16 + l16;
  const float* xrow = x + m * ND;
  int nbase = ng * 64 + l16;

  v8f acc[4] = {};
  for (int k = 0; k < ND; k += 4) {
    int kk = k + kh * 2;
    v2f a; a.x = xrow[kk]; a.y = xrow[kk + 1];
    const float* wp = W1 + kk * NH + nbase;
#pragma unroll
    for (int t = 0; t < 4; ++t) {
      v2f b; b.x = wp[t * 16]; b.y = wp[NH + t * 16];
      acc[t] = __builtin_amdgcn_wmma_f32_16x16x4_f32(
          false, a, false, b, (short)0, acc[t], false, false);
    }
  }
#pragma unroll
  for (int t = 0; t < 4; ++t) {
    int n = ng * 64 + t * 16 + l16;
    float bv = bias1[n];
#pragma unroll
    for (int r = 0; r < 8; ++r) {
      int row = mt * 16 + r + 8 * kh;
      float v = acc[t][r] + bv;
      h[row * NH + n] = v > 0.0f ? v : 0.0f;
    }
  }
}

// ---------------- GEMM2: logits = h @ W2 + b2  [128x1024 @ 1024x1000] -----
__global__ void k_gemm2(const float* __restrict__ h, const float* __restrict__ W2,
                        const float* __restrict__ bias2, float* __restrict__ logits) {
  int w    = blockIdx.x * (blockDim.x >> 5) + (threadIdx.x >> 5);
  int lane = threadIdx.x & 31;
  int kh = lane >> 4, l16 = lane & 15;
  int mt = w >> 4, ng = w & 15;
  int m = mt * 16 + l16;
  const float* hrow = h + m * NH;

  v8f acc[4] = {};
  for (int k = 0; k < NH; k += 4) {
    int kk = k + kh * 2;
    v2f a; a.x = hrow[kk]; a.y = hrow[kk + 1];
#pragma unroll
    for (int t = 0; t < 4; ++t) {
      int n = ng * 64 + t * 16 + l16;
      bool ok = (n < NC);
      v2f b;
      b.x = ok ? W2[kk * NC + n] : 0.0f;
      b.y = ok ? W2[(kk + 1) * NC + n] : 0.0f;
      acc[t] = __builtin_amdgcn_wmma_f32_16x16x4_f32(
          false, a, false, b, (short)0, acc[t], false, false);
    }
  }
#pragma unroll
  for (int t = 0; t < 4; ++t) {
    int n = ng * 64 + t * 16 + l16;
    if (n < NC) {
      float bv = bias2[n];
#pragma unroll
      for (int r = 0; r < 8; ++r) {
        int row = mt * 16 + r + 8 * kh;
        logits[row * NCP + n] = acc[t][r] + bv;
      }
    }
  }
}

// ---------------- sample y_i, dlogits = softmax - onehot(y) ---------------
// one wave (32 threads) per sample
__global__ void k_sample(const float* __restrict__ logits, float* __restrict__ dl) {
  int i = blockIdx.x;
  int lane = threadIdx.x;
  const float* lrow = logits + i * NCP;

  // row max
  float mx = -__builtin_inff();
  for (int j = lane; j < NC; j += 32) mx = fmaxf(mx, lrow[j]);
  for (int s = 16; s > 0; s >>= 1) mx = fmaxf(mx, __shfl_xor(mx, s, 32));
  // sum exp
  float z = 0.0f;
  for (int j = lane; j < NC; j += 32) z += expf(lrow[j] - mx);
  for (int s = 16; s > 0; s >>= 1) z += __shfl_xor(z, s, 32);

  // key_i = threefry((0,42), (i, 128+i))  (jax.random.split(key(42), 128))
  unsigned k0, k1;
  threefry2x32(0u, 42u, (unsigned)i, (unsigned)(NB + i), k0, k1);

  // gumbel-argmax:  bits[j], bits[j+500] = threefry(key_i, (j, j+500))
  float best = -__builtin_inff();
  int bidx = 0x7fffffff;
  for (int j = lane; j < NC / 2; j += 32) {
    unsigned b0, b1;
    threefry2x32(k0, k1, (unsigned)j, (unsigned)(j + NC / 2), b0, b1);
    float c0 = lrow[j] + gumbel_from_bits(b0);
    float c1 = lrow[j + NC / 2] + gumbel_from_bits(b1);
    if (c0 > best || (c0 == best && j < bidx)) { best = c0; bidx = j; }
    if (c1 > best || (c1 == best && (j + NC / 2) < bidx)) { best = c1; bidx = j + NC / 2; }
  }
  for (int s = 16; s > 0; s >>= 1) {
    float ov = __shfl_xor(best, s, 32);
    int   oi = __shfl_xor(bidx, s, 32);
    if (ov > best || (ov == best && oi < bidx)) { best = ov; bidx = oi; }
  }

  float* drow = dl + i * NCP;
  for (int j = lane; j < NCP; j += 32) {
    float v = 0.0f;
    if (j < NC) v = expf(lrow[j] - mx) / z - (j == bidx ? 1.0f : 0.0f);
    drow[j] = v;  // zero padding in cols 1000..1023
  }
}

// ------- GEMM3: dhpre = (dlogits @ W2^T) * (h > 0)  [128x1000 @ 1000x1024] -
__global__ void k_gemm3(const float* __restrict__ dl, const float* __restrict__ W2,
                        const float* __restrict__ h, float* __restrict__ dhp) {
  int w    = blockIdx.x * (blockDim.x >> 5) + (threadIdx.x >> 5);
  int lane = threadIdx.x & 31;
  int kh = lane >> 4, l16 = lane & 15;
  int mt = w >> 4, ng = w & 15;
  int m = mt * 16 + l16;
  const float* arow = dl + m * NCP;
  int nbase = ng * 64 + l16;

  v8f acc[4] = {};
  for (int k = 0; k < NC; k += 4) {   // 1000 = 4*250
    int kk = k + kh * 2;
    v2f a; a.x = arow[kk]; a.y = arow[kk + 1];
#pragma unroll
    for (int t = 0; t < 4; ++t) {
      int n = nbase + t * 16;
      const float* wp = W2 + n * NC + kk;  // B[k][n] = W2[n][k]
      v2f b; b.x = wp[0]; b.y = wp[1];
      acc[t] = __builtin_amdgcn_wmma_f32_16x16x4_f32(
          false, a, false, b, (short)0, acc[t], false, false);
    }
  }
#pragma unroll
  for (int t = 0; t < 4; ++t) {
    int n = ng * 64 + t * 16 + l16;
#pragma unroll
    for (int r = 0; r < 8; ++r) {
      int row = mt * 16 + r + 8 * kh;
      float v = acc[t][r];
      dhp[row * NH + n] = (h[row * NH + n] > 0.0f) ? v : 0.0f;
    }
  }
}

// ---------------- final: row norms -> 5 traces ----------------------------
__global__ void k_final(const float* __restrict__ x, const float* __restrict__ h,
                        const float* __restrict__ dl, const float* __restrict__ dhp,
                        float* __restrict__ out) {
  int i = threadIdx.x;  // one sample per thread (128 threads)
  float xs = 0.f, hs = 0.f, ds = 0.f, dls = 0.f;
  for (int d = 0; d < ND; ++d)  { float v = x[i * ND + d];   xs  += v * v; }
  for (int j = 0; j < NH; ++j)  { float v = h[i * NH + j];   hs  += v * v; }
  for (int j = 0; j < NH; ++j)  { float v = dhp[i * NH + j]; ds  += v * v; }
  for (int c = 0; c < NCP; ++c) { float v = dl[i * NCP + c]; dls += v * v; }

  __shared__ float s0[NB], s1[NB], s2[NB], s3[NB];
  s0[i] = xs * ds;   // W1 trace contribution
  s1[i] = ds;        // b1
  s2[i] = hs * dls;  // W2
  s3[i] = dls;       // b2
  __syncthreads();
  for (int stride = NB / 2; stride > 0; stride >>= 1) {
    if (i < stride) {
      s0[i] += s0[i + stride];
      s1[i] += s1[i + stride];
      s2[i] += s2[i + stride];
      s3[i] += s3[i + stride];
    }
    __syncthreads();
  }
  if (i == 0) {
    const float inv = 1.0f / (float)NB;
    float tW1 = s0[0] * inv, tb1 = s1[0] * inv, tW2 = s2[0] * inv, tb2 = s3[0] * inv;
    out[0] = tW1 + tb1 + tW2 + tb2;
    out[1] = tW1;
    out[2] = tb1;
    out[3] = tW2;
    out[4] = tb2;
  }
}

extern "C" void kernel_launch(void* const* d_in, const int* in_sizes, int n_in,
                              void* d_out, int out_size, void* d_ws, size_t ws_size,
                              hipStream_t stream) {
  const float* x   = (const float*)d_in[0];
  const float* W1  = (const float*)d_in[1];
  const float* b1  = (const float*)d_in[2];
  const float* W2  = (const float*)d_in[3];
  const float* b2  = (const float*)d_in[4];
  float* out = (float*)d_out;

  float* ws     = (float*)d_ws;
  float* h      = ws;                  // 128*1024
  float* logits = ws + 1 * NB * NCP;   // 128*1024
  float* dl     = ws + 2 * NB * NCP;   // 128*1024
  float* dhp    = ws + 3 * NB * NCP;   // 128*1024   (2 MB total)

  // 128 waves per GEMM: 32 blocks x 128 threads (4 waves/block)
  k_gemm1 <<<32, 128, 0, stream>>>(x, W1, b1, h);
  k_gemm2 <<<32, 128, 0, stream>>>(h, W2, b2, logits);
  k_sample<<<NB, 32, 0, stream>>>(logits, dl);
  k_gemm3 <<<32, 128, 0, stream>>>(dl, W2, h, dhp);
  k_final <<<1, NB, 0, stream>>>(x, h, dl, dhp, out);
}